// EMD_71004399337519
// MI455X (gfx1250) — compile-verified
//
#include <hip/hip_runtime.h>

typedef __attribute__((ext_vector_type(2))) float v2f;
typedef __attribute__((ext_vector_type(8))) float v8f;

#define NSLOT 6
#define DDIM 64
#define BPW 16          // batches per workgroup (8 waves x 2)
#define ROWF (NSLOT*DDIM)   // 384 floats per batch per matrix

__global__ __launch_bounds__(256) void emd_kernel(const float* __restrict__ preds,
                                                  const float* __restrict__ targets,
                                                  float* __restrict__ partial) {
    __shared__ alignas(16) float lds_p[BPW * ROWF];   // 24 KB
    __shared__ alignas(16) float lds_t[BPW * ROWF];   // 24 KB
    __shared__ float lds_dot[BPW * 40];               // 6x6 per batch, padded stride 40
    __shared__ float lds_nred[256];
    __shared__ float lds_dp[BPW * 64];
    __shared__ float lds_best[BPW];

    const int tid = threadIdx.x;
    const size_t g0 = (size_t)blockIdx.x * BPW;

    // ---- Stage 1: fully coalesced float4 copy of two contiguous 24KB chunks ----
    {
        const float4* gp4 = (const float4*)(preds + g0 * ROWF);
        const float4* gt4 = (const float4*)(targets + g0 * ROWF);
        float4* lp4 = (float4*)lds_p;
        float4* lt4 = (float4*)lds_t;
#pragma unroll
        for (int it = 0; it < 6; ++it) {          // 1536 float4 per matrix
            int i = tid + it * 256;
            lp4[i] = gp4[i];
            lt4[i] = gt4[i];
        }
    }
    __syncthreads();

    const int lane = tid & 31;
    const int wv   = tid >> 5;     // wave id 0..7
    const int half = lane >> 4;    // lane group
    const int l16  = lane & 15;
    const int lb0  = 2 * wv;       // local batch of this wave's tile

    // A-matrix: lane supplies row m = l16; K-pair offset 2*half.
    // B-matrix: lane supplies col n = l16; K-row pair offset 2*half.
    // Rows/cols 0..5 -> batch lb0, 8..13 -> batch lb0+1; 6,7,14,15 -> zero.
    const int m     = l16;
    const int sub   = m & 7;
    const bool vld  = (sub < NSLOT);
    const int subc  = vld ? sub : 0;               // safe address
    const int lb    = lb0 + (m >> 3);
    const float* arow = &lds_p[(lb * NSLOT + subc) * DDIM];
    const float* brow = &lds_t[(lb * NSLOT + subc) * DDIM];
    const int koff = half * 2;

    v8f c = {0.f, 0.f, 0.f, 0.f, 0.f, 0.f, 0.f, 0.f};
#pragma unroll
    for (int kk = 0; kk < 16; ++kk) {              // K = 64 in steps of 4
        v2f av = *(const v2f*)(arow + 4 * kk + koff);
        v2f bv = *(const v2f*)(brow + 4 * kk + koff);
        v2f a, b;
        a.x = vld ? av.x : 0.f;  a.y = vld ? av.y : 0.f;   // v_cndmask, EXEC stays full
        b.x = vld ? bv.x : 0.f;  b.y = vld ? bv.y : 0.f;
        c = __builtin_amdgcn_wmma_f32_16x16x4_f32(false, a, false, b,
                                                  (short)0, c, false, false);
    }

    // ---- Per-lane norm partials: batch lb0+half, lanes l16 stride 16 ----
    {
        const int nlb = lb0 + half;
        const float* pB = &lds_p[nlb * ROWF];
        const float* tB = &lds_t[nlb * ROWF];
        float acc = 0.f;
#pragma unroll
        for (int j = 0; j < 24; ++j) {
            float x = pB[l16 + 16 * j]; acc = fmaf(x, x, acc);
            float y = tB[l16 + 16 * j]; acc = fmaf(y, y, acc);
        }
        lds_nred[tid] = acc;
    }

    // ---- Extract the two 6x6 dot blocks from the accumulator ----
    // lane L, vgpr v holds (M = v + 8*half, N = l16).
    {
        bool w0 = (half == 0) && (l16 < 6);                 // batch0 block
        bool w1 = (half == 1) && (l16 >= 8) && (l16 < 14);  // batch1 block
        if (w0 | w1) {
            int col = l16 - 8 * half;                       // 0..5
            float* dst = &lds_dot[(lb0 + half) * 40 + col * 6];
            dst[0] = c[0]; dst[1] = c[1]; dst[2] = c[2];
            dst[3] = c[3]; dst[4] = c[4]; dst[5] = c[5];
        }
    }
    __syncthreads();

    // ---- Bitmask-DP max assignment + norm finish: one lane per batch ----
    if (tid < BPW) {
        const float* nr = &lds_nred[(tid >> 1) * 32 + (tid & 1) * 16];
        float nsum = 0.f;
#pragma unroll
        for (int i = 0; i < 16; ++i) nsum += nr[i];

        float* dp = &lds_dp[tid * 64];
        const float* dm = &lds_dot[tid * 40];   // dm[col*6 + row]
        dp[0] = 0.f;
        for (int mask = 1; mask < 64; ++mask) {
            int r = __popc(mask) - 1;           // pred row index
            float best = -3.0e38f;
            for (int j = 0; j < 6; ++j) {
                int bit = 1 << j;
                if (mask & bit) {
                    float v = dp[mask ^ bit] + dm[j * 6 + r];
                    best = fmaxf(best, v);
                }
            }
            dp[mask] = best;
        }
        // min-cost = sum of norms - 2 * max assignment
        lds_best[tid] = nsum - 2.0f * dp[63];
    }
    __syncthreads();

    if (tid == 0) {
        float s = 0.f;
#pragma unroll
        for (int i = 0; i < BPW; ++i) s += lds_best[i];
        partial[blockIdx.x] = s;
    }
}

__global__ __launch_bounds__(256) void reduce_kernel(const float* __restrict__ partial,
                                                     int n, float* __restrict__ out) {
    __shared__ float sh[256];
    int tid = threadIdx.x;
    float s = 0.f;
    for (int i = tid; i < n; i += 256) s += partial[i];   // fixed order -> deterministic
    sh[tid] = s;
    __syncthreads();
    for (int off = 128; off > 0; off >>= 1) {
        if (tid < off) sh[tid] += sh[tid + off];
        __syncthreads();
    }
    if (tid == 0) out[0] = sh[0];
}

extern "C" void kernel_launch(void* const* d_in, const int* in_sizes, int n_in,
                              void* d_out, int out_size, void* d_ws, size_t ws_size,
                              hipStream_t stream) {
    const float* preds   = (const float*)d_in[0];
    const float* targets = (const float*)d_in[1];
    int B = in_sizes[0] / (NSLOT * DDIM);   // 32768
    int nblocks = B / BPW;                  // 2048
    float* partial = (float*)d_ws;          // 8 KB scratch

    emd_kernel<<<nblocks, 256, 0, stream>>>(preds, targets, partial);
    reduce_kernel<<<1, 256, 0, stream>>>(partial, nblocks, (float*)d_out);
}